// NMS_46574625357925
// MI455X (gfx1250) — compile-verified
//
#include <hip/hip_runtime.h>
#include <hip/hip_bf16.h>

typedef __attribute__((ext_vector_type(2))) float v2f;
typedef __attribute__((ext_vector_type(8))) float v8f;

#define NMS_N      6144
#define NMS_TILES  (NMS_N / 16)     // 384
#define NMS_WORDS  (NMS_N / 32)     // 192
#define IOU_T      0.5f

// ---------------------------------------------------------------------------
// 1) Bitonic sort of (conf desc, idx asc) in one workgroup. 8192 slots in LDS.
// ---------------------------------------------------------------------------
__global__ void nms_sort_kernel(const float* __restrict__ boxes,
                                unsigned* __restrict__ order, int N) {
  __shared__ unsigned long long keys[8192];
  const int tid = threadIdx.x;  // 1024 threads
  for (int i = tid; i < 8192; i += 1024) {
    unsigned long long k;
    if (i < N) {
      float conf = boxes[i * 7 + 5];                  // conf in [0,1)
      unsigned cb = __float_as_uint(conf);            // order-preserving for >=0
      k = ((unsigned long long)cb << 32) | (unsigned)(0xFFFFFFFFu - (unsigned)i);
    } else {
      k = 0ull;                                       // pads sort to the end
    }
    keys[i] = ~k;  // ascending sort on ~k == descending on k
  }
  __syncthreads();
  for (int kk = 2; kk <= 8192; kk <<= 1) {
    for (int j = kk >> 1; j > 0; j >>= 1) {
      for (int p = tid; p < 4096; p += 1024) {
        int i = ((p & ~(j - 1)) << 1) | (p & (j - 1));
        int ixj = i | j;
        unsigned long long a = keys[i], b = keys[ixj];
        bool up = ((i & kk) == 0);
        if ((a > b) == up) { keys[i] = b; keys[ixj] = a; }
      }
      __syncthreads();
    }
  }
  for (int i = tid; i < N; i += 1024) {
    unsigned long long k = ~keys[i];
    order[i] = 0xFFFFFFFFu - (unsigned)(k & 0xFFFFFFFFu);
  }
}

// ---------------------------------------------------------------------------
// 2) Gather sorted SoA: x1,y1,x2,y2,area,|c|^2,cls,cx,cy  (9 arrays, stride N)
// ---------------------------------------------------------------------------
__global__ void nms_gather_kernel(const float* __restrict__ boxes,
                                  const unsigned* __restrict__ order,
                                  float* __restrict__ S, int N) {
  int i = blockIdx.x * blockDim.x + threadIdx.x;
  if (i >= N) return;
  unsigned o = order[i];
  float cx = boxes[o * 7 + 1], cy = boxes[o * 7 + 2];
  float w  = boxes[o * 7 + 3], h  = boxes[o * 7 + 4];
  float cl = boxes[o * 7 + 6];
  float x1 = cx - 0.5f * w, y1 = cy - 0.5f * h;
  float x2 = cx + 0.5f * w, y2 = cy + 0.5f * h;
  float area = (x2 - x1) * (y2 - y1);
  float ccx = (x1 + x2) * 0.5f, ccy = (y1 + y2) * 0.5f;
  S[0 * N + i] = x1;  S[1 * N + i] = y1;
  S[2 * N + i] = x2;  S[3 * N + i] = y2;
  S[4 * N + i] = area;
  S[5 * N + i] = ccx * ccx + ccy * ccy;
  S[6 * N + i] = cl;
  S[7 * N + i] = ccx; S[8 * N + i] = ccy;
}

// ---------------------------------------------------------------------------
// 3) Suppression bitmask. One block per 16-row tile (8 waves); each wave owns
//    mask words w = rt/2 + waveId step 8 (balanced, jointly covering
//    [rt/2,192)). WMMA f32 16x16x4 computes the center cross term.
//    Division-free test: diou > T  <=>  inter*c2 > uni*(T*c2 + d2).
// ---------------------------------------------------------------------------
__global__ void __launch_bounds__(256)
nms_mask_kernel(const float* __restrict__ S, unsigned* __restrict__ maskBuf,
                int N) {
  __shared__ float rowLds[16][8];
  const int wave = threadIdx.x >> 5;
  const int lane = threadIdx.x & 31;
  const int rt = blockIdx.x;                    // row tile (384 blocks)
  const int rBase = rt * 16;
  const bool hi = lane >= 16;
  const int m = lane & 15;                      // col-in-tile / row-owned

  const float* SX1 = S + 0 * N; const float* SY1 = S + 1 * N;
  const float* SX2 = S + 2 * N; const float* SY2 = S + 3 * N;
  const float* SAR = S + 4 * N; const float* SSQ = S + 5 * N;
  const float* SCL = S + 6 * N; const float* SCX = S + 7 * N;
  const float* SCY = S + 8 * N;

  if (threadIdx.x < 16) {
    int r = rBase + threadIdx.x;
    rowLds[threadIdx.x][0] = SX1[r];
    rowLds[threadIdx.x][1] = SY1[r];
    rowLds[threadIdx.x][2] = SX2[r];
    rowLds[threadIdx.x][3] = SY2[r];
    rowLds[threadIdx.x][4] = SAR[r];
    rowLds[threadIdx.x][5] = SSQ[r];
    rowLds[threadIdx.x][6] = SCL[r];
  }
  __syncthreads();  // uniform: executed by all waves before any divergence

  // 0/1 lane mask (hoisted) so operand loads stay unconditional (no exec
  // branches): lanes 16-31 feed K=2,3 which must be zero.
  const float lmask = hi ? 0.0f : 1.0f;

  // A operand: 16x4 f32, lanes0-15 hold K=0,1 (cx,cy) of rows 0..15.
  v2f aV;
  aV.x = SCX[rBase + m] * lmask;
  aV.y = SCY[rBase + m] * lmask;

  for (int w = (rt >> 1) + wave; w < NMS_WORDS; w += 8) {
    unsigned word = 0;
#pragma unroll
    for (int half = 0; half < 2; ++half) {
      const int cBase = (w * 2 + half) * 16;
      const int col = cBase + m;
      const float x1c = SX1[col], y1c = SY1[col];
      const float x2c = SX2[col], y2c = SY2[col];
      const float arC = SAR[col], sqC = SSQ[col], clC = SCL[col];

      v2f bV;
      bV.x = SCX[col] * lmask;
      bV.y = SCY[col] * lmask;
      v8f acc = {};
      // cross[M][Nc] = cxR[M]*cxC[Nc] + cyR[M]*cyC[Nc]
      acc = __builtin_amdgcn_wmma_f32_16x16x4_f32(
          false, aV, false, bV, (short)0, acc, false, false);

      unsigned ball[8];
#pragma unroll
      for (int r = 0; r < 8; ++r) {
        const int mm = r + (hi ? 8 : 0);        // D layout: M=r / r+8
        const float* R = rowLds[mm];
        const float x1r = R[0], y1r = R[1], x2r = R[2], y2r = R[3];
        const float arR = R[4], sqR = R[5], clR = R[6];
        float iw = fminf(x2r, x2c) - fmaxf(x1r, x1c);
        float ih = fminf(y2r, y2c) - fmaxf(y1r, y1c);
        iw = fmaxf(iw, 0.0f);
        ih = fmaxf(ih, 0.0f);
        float inter = iw * ih;
        float uni = (arR + arC) - inter;        // > 0 (boxes non-degenerate)
        float ex = fmaxf(x2r, x2c) - fminf(x1r, x1c);
        float ey = fmaxf(y2r, y2c) - fminf(y1r, y1c);
        float c2 = fmaf(ex, ex, ey * ey);       // > 0
        float d2 = fmaf(-2.0f, acc[r], sqR + sqC);
        // diou > T  <=>  inter*c2 > uni*(T*c2 + d2)
        float lhs = inter * c2;
        float rhs = uni * fmaf(IOU_T, c2, d2);
        const int rowg = rBase + mm, colg = cBase + m;
        bool cond = (clR == clC) && (colg > rowg) && (lhs > rhs);
        ball[r] = __builtin_amdgcn_ballot_w32(cond);
      }
      // Select the 16-bit mask for row m via static select tree.
      unsigned t0 = (m & 1) ? ball[1] : ball[0];
      unsigned t1 = (m & 1) ? ball[3] : ball[2];
      unsigned t2 = (m & 1) ? ball[5] : ball[4];
      unsigned t3 = (m & 1) ? ball[7] : ball[6];
      unsigned u0 = (m & 2) ? t1 : t0;
      unsigned u1 = (m & 2) ? t3 : t2;
      unsigned v  = (m & 4) ? u1 : u0;
      unsigned bits = (m & 8) ? (v >> 16) : (v & 0xFFFFu);
      word |= bits << (half * 16);
    }
    if (lane < 16)
      maskBuf[(size_t)(rBase + m) * NMS_WORDS + w] = word;
  }
}

// ---------------------------------------------------------------------------
// 4) Greedy sequential scan: one wave, removed-bitmask in LDS (192 words).
// ---------------------------------------------------------------------------
__global__ void nms_scan_kernel(const unsigned* __restrict__ maskBuf,
                                unsigned* __restrict__ keepSorted, int N) {
  __shared__ unsigned rem[NMS_WORDS];
  const int lane = threadIdx.x;  // 32 threads
  for (int w = lane; w < NMS_WORDS; w += 32) rem[w] = 0u;
  __syncthreads();
  for (int i = 0; i < N; ++i) {
    unsigned wv = rem[i >> 5];
    bool kept = ((wv >> (i & 31)) & 1u) == 0u;
    if (kept) {
      const unsigned* row = maskBuf + (size_t)i * NMS_WORDS;
      for (int w = (i >> 5) + lane; w < NMS_WORDS; w += 32) rem[w] |= row[w];
    }
    if (lane == 0) keepSorted[i] = kept ? 1u : 0u;
    __syncthreads();
  }
}

// ---------------------------------------------------------------------------
// 5) Scatter: out[orig] = boxes[orig] * keep
// ---------------------------------------------------------------------------
__global__ void nms_final_kernel(const float* __restrict__ boxes,
                                 const unsigned* __restrict__ order,
                                 const unsigned* __restrict__ keepSorted,
                                 float* __restrict__ out, int N) {
  int i = blockIdx.x * blockDim.x + threadIdx.x;
  if (i >= N) return;
  unsigned o = order[i];
  float k = keepSorted[i] ? 1.0f : 0.0f;
#pragma unroll
  for (int c = 0; c < 7; ++c) out[o * 7 + c] = boxes[o * 7 + c] * k;
}

extern "C" void kernel_launch(void* const* d_in, const int* in_sizes, int n_in,
                              void* d_out, int out_size, void* d_ws,
                              size_t ws_size, hipStream_t stream) {
  const float* boxes = (const float*)d_in[0];
  float* out = (float*)d_out;
  const int N = in_sizes[0] / 7;  // 6144

  size_t off = 0;
  auto alloc = [&](size_t bytes) -> void* {
    void* p = (char*)d_ws + off;
    off += (bytes + 255) & ~(size_t)255;
    return p;
  };
  unsigned* order    = (unsigned*)alloc((size_t)N * 4);
  unsigned* keepS    = (unsigned*)alloc((size_t)N * 4);
  float*    S        = (float*)   alloc((size_t)9 * N * 4);
  unsigned* maskBuf  = (unsigned*)alloc((size_t)N * NMS_WORDS * 4);
  (void)ws_size;

  nms_sort_kernel<<<1, 1024, 0, stream>>>(boxes, order, N);
  nms_gather_kernel<<<(N + 255) / 256, 256, 0, stream>>>(boxes, order, S, N);
  nms_mask_kernel<<<NMS_TILES, 256, 0, stream>>>(S, maskBuf, N);
  nms_scan_kernel<<<1, 32, 0, stream>>>(maskBuf, keepS, N);
  nms_final_kernel<<<(N + 255) / 256, 256, 0, stream>>>(boxes, order, keepS,
                                                        out, N);
}